// SingleGAT_29025388987030
// MI455X (gfx1250) — compile-verified
//
#include <hip/hip_runtime.h>
#include <cstdint>

typedef __attribute__((ext_vector_type(2))) float v2f;
typedef __attribute__((ext_vector_type(8))) float v8f;

#define IN_C   128
#define HC     128   // H * OUT_C
#define NHEAD  4
#define OUT_C  32
#define NEG_SLOPE 0.2f

// ---- ordered-float <-> uint key (monotonic under unsigned compare) ----
__device__ __forceinline__ unsigned fkey(float f) {
    unsigned b = __float_as_uint(f);
    return b ^ ((unsigned)((int)b >> 31) | 0x80000000u);
}
__device__ __forceinline__ float fromkey(unsigned k) {
    unsigned b = (k & 0x80000000u) ? (k ^ 0x80000000u) : ~k;
    return __uint_as_float(b);
}
__device__ __forceinline__ float leaky(float v) { return v > 0.0f ? v : NEG_SLOPE * v; }

// ------------------------------------------------------------------
// Kernel 1: h = x @ W   ([N,128] @ [128,128]) via V_WMMA_F32_16X16X4_F32
// One wave computes one 16x16 tile; K loop of 32 WMMA ops.
// A layout (16x4 f32): lane L -> M = L&15, Kpair = (L>>4)*2 (2 VGPRs)
// B layout (4x16 f32): lane L -> Ncol = L&15, Kpair = (L>>4)*2
// C/D (16x16 f32): VGPR r -> M = r + 8*(L>>4), N = L&15
// ------------------------------------------------------------------
__global__ void gemm_wmma_kernel(const float* __restrict__ x,
                                 const float* __restrict__ W,
                                 float* __restrict__ h, int N, int mtiles) {
    int wave = (int)((blockIdx.x * blockDim.x + threadIdx.x) >> 5);
    int lane = threadIdx.x & 31;
    const int ntiles = HC / 16;                 // 8 column tiles
    int tm = wave / ntiles;
    int tn = wave - tm * ntiles;
    if (tm >= mtiles) return;                   // wave-uniform (no WMMA divergence)

    int m     = lane & 15;                      // row (A) / col (B,C)
    int khalf = lane >> 4;                      // 0 or 1

    int rowA = tm * 16 + m;
    int rowAc = rowA < N ? rowA : (N - 1);      // clamp loads for ragged M tile
    const float* __restrict__ Arow = x + (size_t)rowAc * IN_C;
    int colB = tn * 16 + m;

    v8f acc = (v8f)0.0f;
    #pragma unroll
    for (int k = 0; k < IN_C; k += 4) {
        v2f a, b;
        int kk = k + khalf * 2;
        a.x = Arow[kk];
        a.y = Arow[kk + 1];
        b.x = W[(size_t)(kk)     * HC + colB];
        b.y = W[(size_t)(kk + 1) * HC + colB];
        acc = __builtin_amdgcn_wmma_f32_16x16x4_f32(
                false, a, false, b, (short)0, acc, false, false);
    }

    #pragma unroll
    for (int r = 0; r < 8; ++r) {
        int M = tm * 16 + r + 8 * khalf;
        if (M < N) h[(size_t)M * HC + tn * 16 + m] = acc[r];
    }
}

// ------------------------------------------------------------------
// Kernel 2: a_src[n,h] = h[n,h,:] . att_src[h,:]   (same for a_dst)
// ------------------------------------------------------------------
__global__ void att_kernel(const float* __restrict__ h,
                           const float* __restrict__ att_src,
                           const float* __restrict__ att_dst,
                           float* __restrict__ a_src,
                           float* __restrict__ a_dst, int N) {
    int t = blockIdx.x * blockDim.x + threadIdx.x;
    if (t >= N * NHEAD) return;
    int n = t >> 2, hd = t & 3;
    const float* __restrict__ hp = h + (size_t)n * HC + hd * OUT_C;
    const float* __restrict__ as = att_src + hd * OUT_C;
    const float* __restrict__ ad = att_dst + hd * OUT_C;
    float s0 = 0.0f, s1 = 0.0f;
    #pragma unroll
    for (int c = 0; c < OUT_C; ++c) {
        float v = hp[c];
        s0 += v * as[c];
        s1 += v * ad[c];
    }
    a_src[t] = s0;
    a_dst[t] = s1;
}

// ------------------------------------------------------------------
// Kernel 3: init maxkey = key(-inf), denom = 0, out = 0
// ------------------------------------------------------------------
__global__ void init_kernel(unsigned* __restrict__ maxkey,
                            float* __restrict__ denom,
                            float* __restrict__ out, int N) {
    int t = blockIdx.x * blockDim.x + threadIdx.x;
    if (t < N * OUT_C) out[t] = 0.0f;
    if (t < N * NHEAD) {
        maxkey[t] = 0x007FFFFFu;   // fkey(-inf)
        denom[t]  = 0.0f;
    }
}

// ------------------------------------------------------------------
// Kernel 4: per-edge logits -> atomic segment-max (ordered-uint keys)
// float4 loads (global_load_b128) for the per-node head quads.
// ------------------------------------------------------------------
__global__ void edge_max_kernel(const long long* __restrict__ ei,
                                const float4* __restrict__ a_src4,
                                const float4* __restrict__ a_dst4,
                                unsigned* __restrict__ maxkey, int E) {
    int e = blockIdx.x * blockDim.x + threadIdx.x;
    if (e >= E) return;
    int s = (int)ei[e];
    int d = (int)ei[(size_t)E + e];
    float4 as = a_src4[s];
    float4 ad = a_dst4[d];
    atomicMax(&maxkey[d * NHEAD + 0], fkey(leaky(as.x + ad.x)));
    atomicMax(&maxkey[d * NHEAD + 1], fkey(leaky(as.y + ad.y)));
    atomicMax(&maxkey[d * NHEAD + 2], fkey(leaky(as.z + ad.z)));
    atomicMax(&maxkey[d * NHEAD + 3], fkey(leaky(as.w + ad.w)));
}

// ------------------------------------------------------------------
// Kernel 5: exp(logit - segmax) -> atomic segment-sum (denominator)
// ------------------------------------------------------------------
__global__ void edge_sum_kernel(const long long* __restrict__ ei,
                                const float4* __restrict__ a_src4,
                                const float4* __restrict__ a_dst4,
                                const uint4* __restrict__ maxkey4,
                                float* __restrict__ denom, int E) {
    int e = blockIdx.x * blockDim.x + threadIdx.x;
    if (e >= E) return;
    int s = (int)ei[e];
    int d = (int)ei[(size_t)E + e];
    float4 as = a_src4[s];
    float4 ad = a_dst4[d];
    uint4  mk = maxkey4[d];
    atomicAdd(&denom[d * NHEAD + 0], expf(leaky(as.x + ad.x) - fromkey(mk.x)));
    atomicAdd(&denom[d * NHEAD + 1], expf(leaky(as.y + ad.y) - fromkey(mk.y)));
    atomicAdd(&denom[d * NHEAD + 2], expf(leaky(as.z + ad.z) - fromkey(mk.z)));
    atomicAdd(&denom[d * NHEAD + 3], expf(leaky(as.w + ad.w) - fromkey(mk.w)));
}

// ------------------------------------------------------------------
// Kernel 6: wave-per-edge weighted scatter.
// Lane L computes alpha for head (L&3) -- ONE expf / ONE divide per wave
// (SIMD across lanes) -- then the 4 alphas are broadcast via __shfl.
// lane = output channel c; acc = sum_h alpha[h] * h[src,h,c];
// single coalesced atomicAdd into out[dst,c], scaled by 1/NHEAD (head mean).
// ------------------------------------------------------------------
__global__ void edge_scatter_kernel(const long long* __restrict__ ei,
                                    const float* __restrict__ h,
                                    const float* __restrict__ a_src,
                                    const float* __restrict__ a_dst,
                                    const unsigned* __restrict__ maxkey,
                                    const float* __restrict__ denom,
                                    float* __restrict__ out, int E) {
    long long gid = (long long)blockIdx.x * blockDim.x + threadIdx.x;
    int e = (int)(gid >> 5);
    int lane = (int)(gid & 31);
    if (e >= E) return;
    int s = (int)ei[e];
    int d = (int)ei[(size_t)E + e];

    // per-lane head assignment: all loads in-bounds, 1 expf/div per wave
    int hh = lane & 3;
    float l  = leaky(a_src[s * NHEAD + hh] + a_dst[d * NHEAD + hh]);
    float m  = fromkey(maxkey[d * NHEAD + hh]);
    float alpha = expf(l - m) / (denom[d * NHEAD + hh] + 1e-16f);

    const float* __restrict__ hrow = h + (size_t)s * HC;
    float acc = 0.0f;
    #pragma unroll
    for (int hd = 0; hd < NHEAD; ++hd) {
        float a_h = __shfl(alpha, hd, 32);      // alpha for head hd lives in lane hd
        acc += a_h * hrow[hd * OUT_C + lane];
    }
    atomicAdd(&out[(size_t)d * OUT_C + lane], 0.25f * acc);
}

// ------------------------------------------------------------------
// Kernel 7: out = relu(out + bias)
// ------------------------------------------------------------------
__global__ void finalize_kernel(float* __restrict__ out,
                                const float* __restrict__ bias, int N) {
    int t = blockIdx.x * blockDim.x + threadIdx.x;
    if (t >= N * OUT_C) return;
    out[t] = fmaxf(out[t] + bias[t & (OUT_C - 1)], 0.0f);
}

extern "C" void kernel_launch(void* const* d_in, const int* in_sizes, int n_in,
                              void* d_out, int out_size, void* d_ws, size_t ws_size,
                              hipStream_t stream) {
    const float*     x       = (const float*)d_in[0];
    const long long* ei      = (const long long*)d_in[1];   // int64 per reference
    const float*     W       = (const float*)d_in[2];
    const float*     att_src = (const float*)d_in[3];
    const float*     att_dst = (const float*)d_in[4];
    const float*     bias    = (const float*)d_in[5];
    float*           out     = (float*)d_out;

    const int N = in_sizes[0] / IN_C;   // 100000
    const int E = in_sizes[1] / 2;      // 1700000

    // Workspace layout (~57.6 MB, all 16B-aligned): h | a_src | a_dst | maxkey | denom
    char*  ws      = (char*)d_ws;
    float* h       = (float*)ws;                       size_t off = (size_t)N * HC * sizeof(float);
    float* a_src   = (float*)(ws + off);               off += (size_t)N * NHEAD * sizeof(float);
    float* a_dst   = (float*)(ws + off);               off += (size_t)N * NHEAD * sizeof(float);
    unsigned* mkey = (unsigned*)(ws + off);            off += (size_t)N * NHEAD * sizeof(unsigned);
    float* denom   = (float*)(ws + off);

    const int mtiles = (N + 15) / 16;
    const int waves  = mtiles * (HC / 16);             // 50000 waves, 1 tile each

    gemm_wmma_kernel<<<(waves + 7) / 8, 256, 0, stream>>>(x, W, h, N, mtiles);

    att_kernel<<<(N * NHEAD + 255) / 256, 256, 0, stream>>>(h, att_src, att_dst,
                                                            a_src, a_dst, N);

    init_kernel<<<(N * OUT_C + 255) / 256, 256, 0, stream>>>(mkey, denom, out, N);

    edge_max_kernel<<<(E + 255) / 256, 256, 0, stream>>>(
        ei, (const float4*)a_src, (const float4*)a_dst, mkey, E);

    edge_sum_kernel<<<(E + 255) / 256, 256, 0, stream>>>(
        ei, (const float4*)a_src, (const float4*)a_dst, (const uint4*)mkey, denom, E);

    long long scatter_threads = (long long)E * 32;
    edge_scatter_kernel<<<(unsigned)((scatter_threads + 255) / 256), 256, 0, stream>>>(
        ei, h, a_src, a_dst, mkey, denom, out, E);

    finalize_kernel<<<(N * OUT_C + 255) / 256, 256, 0, stream>>>(out, bias, N);
}